// Net_88081189307142
// MI455X (gfx1250) — compile-verified
//
#include <hip/hip_runtime.h>

// CDNA5 / gfx1250, wave32. Fused SNN forward:
//   cur1 = x@W1^T + b1        -> 2x v_wmma_f32_16x16x4_f32 (K=3 padded to 4)
//   100-step LIF recurrence   -> membrane state held in 16 VGPRs (C/D layout)
//   cur2 = spk@W2^T + b2      -> 8x v_wmma_f32_16x16x4_f32 (K=32)
typedef float v2f __attribute__((ext_vector_type(2)));
typedef float v8f __attribute__((ext_vector_type(8)));

#define BETA    0.9f
#define NSTEPS  100
#define LDS_STR 34   // 16-row spike tile, padded stride (floats) for LDS banking

__global__ __launch_bounds__(256) void snn_fused_kernel(
    const float* __restrict__ x,
    const float* __restrict__ W1,
    const float* __restrict__ b1v,
    const float* __restrict__ W2,
    const float* __restrict__ b2v,
    float* __restrict__ out)
{
    __shared__ float spkbuf[8 * 16 * LDS_STR];   // 8 waves x [16][34] f32

    const int lane = threadIdx.x & 31;
    const int wave = threadIdx.x >> 5;
    const int h    = lane >> 4;      // half-wave (selects K/M upper half per WMMA layouts)
    const int ln   = lane & 15;
    const int rowBase = blockIdx.x * 128 + wave * 16;   // 16 batch rows per wave

    // ---------------- GEMM 1: cur1 = x @ W1^T  (M=16 rows, N=32 hidden, K=3->4) -----
    // A-layout 16x4 f32: vgpr0 = K{0,2}, vgpr1 = K{1,3}; lanes 0-15 M=lane, 16-31 M=lane-16
    const int arow = rowBase + ln;
    float ax1 = x[arow * 3 + 1];               // always in-bounds; selected below
    v2f ax;
    ax.x = x[arow * 3 + 2 * h];                // k = 0 or 2
    ax.y = (h == 0) ? ax1 : 0.0f;              // k = 1, k = 3 is zero pad

    // B-layout 4x16 f32 (transposed analog of A): vgpr0 = K{0,2}, vgpr1 = K{1,3}, lane%16 = N
    v2f bw0, bw1;
    {
        int n0 = ln, n1 = ln + 16;
        float t0 = W1[n0 * 3 + 1], t1 = W1[n1 * 3 + 1];
        bw0.x = W1[n0 * 3 + 2 * h];  bw0.y = (h == 0) ? t0 : 0.0f;
        bw1.x = W1[n1 * 3 + 2 * h];  bw1.y = (h == 0) ? t1 : 0.0f;
    }

    v8f zero = {};
    v8f cur0 = __builtin_amdgcn_wmma_f32_16x16x4_f32(false, ax, false, bw0, (short)0, zero, false, false);
    v8f cur1 = __builtin_amdgcn_wmma_f32_16x16x4_f32(false, ax, false, bw1, (short)0, zero, false, false);

    // C/D layout: column N = lane%16 is constant across the 8 accumulator VGPRs.
    const float bias0 = b1v[ln];
    const float bias1 = b1v[ln + 16];

    // ---------------- 100-step LIF recurrence, fully register-resident ---------------
    float c0[8], c1[8], cm0[8], cm1[8], m0[8], m1[8];
#pragma unroll
    for (int r = 0; r < 8; ++r) {
        c0[r] = cur0[r] + bias0;  c1[r] = cur1[r] + bias1;
        cm0[r] = c0[r] - 1.0f;    cm1[r] = c1[r] - 1.0f;   // cur1 - reset*threshold, prefused
        m0[r] = 0.0f;             m1[r] = 0.0f;
    }
#pragma unroll 1
    for (int s = 0; s < NSTEPS; ++s) {
#pragma unroll
        for (int r = 0; r < 8; ++r) {
            // mem = beta*mem + cur1 - H(mem-1)  ==  fma(beta, mem, mem>1 ? cur1-1 : cur1)
            m0[r] = fmaf(BETA, m0[r], (m0[r] > 1.0f) ? cm0[r] : c0[r]);
            m1[r] = fmaf(BETA, m1[r], (m1[r] > 1.0f) ? cm1[r] : c1[r]);
        }
    }

    // ---------------- spikes -> LDS (C/D layout -> canonical [m][k]) ------------------
    const int wbase = wave * 16 * LDS_STR;
#pragma unroll
    for (int r = 0; r < 8; ++r) {
        int m = r + 8 * h;                      // C-layout row of this vgpr
        spkbuf[wbase + m * LDS_STR + ln]      = (m0[r] > 1.0f) ? 1.0f : 0.0f;
        spkbuf[wbase + m * LDS_STR + ln + 16] = (m1[r] > 1.0f) ? 1.0f : 0.0f;
    }
    __syncthreads();   // waves only read their own region; barrier also orders DS ops

    // ---------------- GEMM 2: cur2 = spk @ W2^T  (M=16, N=16 (3 valid), K=32) ---------
    v8f d = {};
    const bool valid = (ln < 3);
    const int  nn    = valid ? ln : 0;          // clamp so W2 reads stay in-bounds
#pragma unroll
    for (int c = 0; c < 8; ++c) {
        int k0 = 4 * c + 2 * h;
        // A chunk: spk[m][k0], spk[m][k0+1]  (8B-aligned: LDS_STR even, k0 even)
        v2f aspk = *reinterpret_cast<const v2f*>(&spkbuf[wbase + ln * LDS_STR + k0]);
        // B chunk: W2^T[k][n] = W2[n][k], zero-padded for n >= 3
        float w0 = W2[nn * 32 + k0];
        float w1 = W2[nn * 32 + k0 + 1];
        v2f bw2;
        bw2.x = valid ? w0 : 0.0f;
        bw2.y = valid ? w1 : 0.0f;
        d = __builtin_amdgcn_wmma_f32_16x16x4_f32(false, aspk, false, bw2, (short)0, d, false, false);
    }

    // ---------------- store cur2 (+b2): only N = 0..2 columns are real -----------------
    if (valid) {
        float bias = b2v[ln];
#pragma unroll
        for (int r = 0; r < 8; ++r) {
            int row = rowBase + r + 8 * h;
            out[row * 3 + ln] = d[r] + bias;
        }
    }
}

extern "C" void kernel_launch(void* const* d_in, const int* in_sizes, int n_in,
                              void* d_out, int out_size, void* d_ws, size_t ws_size,
                              hipStream_t stream) {
    const float* x  = (const float*)d_in[0];   // [B,3]
    const float* W1 = (const float*)d_in[1];   // [32,3]
    const float* b1 = (const float*)d_in[2];   // [32]
    const float* W2 = (const float*)d_in[3];   // [3,32]
    const float* b2 = (const float*)d_in[4];   // [3]
    float* out = (float*)d_out;                // [B,3]

    const int B = in_sizes[0] / 3;             // 65536
    dim3 block(256);                           // 8 wave32s, each owns 16 rows
    dim3 grid(B / 128);                        // 512 blocks
    snn_fused_kernel<<<grid, block, 0, stream>>>(x, W1, b1, W2, b2, out);
}